// Stage1Assigner_36807869726944
// MI455X (gfx1250) — compile-verified
//
#include <hip/hip_runtime.h>
#include <stdint.h>

// Stage1Assigner for MI455X (gfx1250, wave32).
// No matmul structure -> WMMA inapplicable; CDNA5 paths used instead:
//   * global_load_async_to_lds_b128 + s_wait_asynccnt (GT/hq staging into LDS)
//   * global_prefetch_b8 (streaming anchor pass)
// All IoU arithmetic runs with fp contract(off) so recomputed IoUs are
// bitwise identical across kernels (needed for the iou==hq equality rule).

namespace {
constexpr int B = 8;
constexpr int A = 200000;
constexpr int G = 64;
constexpr int K = 4;
constexpr int MAX_POS = 128;   // NUM_SAMPLES * POS_FRAC
constexpr float T_HIGH = 0.7f;
constexpr int THREADS = 256;
constexpr int ACH = (A + THREADS - 1) / THREADS;  // 782 anchor chunks
}  // namespace

// ---------------- box math (bitwise-reproducible) ----------------

__device__ __forceinline__ float4 to_xyxy(float4 b) {
#pragma clang fp contract(off)
  float4 r;
  r.x = b.x - 0.5f * b.z;
  r.y = b.y - 0.5f * b.w;
  r.z = b.x + 0.5f * b.z;
  r.w = b.y + 0.5f * b.w;
  return r;
}

__device__ __forceinline__ float iou_f(float4 g, float4 a) {
#pragma clang fp contract(off)
  float area1 = (g.z - g.x) * (g.w - g.y);
  float area2 = (a.z - a.x) * (a.w - a.y);
  float ltx = fmaxf(g.x, a.x);
  float lty = fmaxf(g.y, a.y);
  float rbx = fminf(g.z, a.z);
  float rby = fminf(g.w, a.w);
  float w = fmaxf(rbx - ltx, 0.0f);
  float h = fmaxf(rby - lty, 0.0f);
  float inter = w * h;
  float uni = (area1 + area2) - inter;
  return inter / uni;
}

// ---------------- JAX threefry2x32 (exact) ----------------

__device__ __forceinline__ uint32_t rotl32(uint32_t x, int d) {
  return (x << d) | (x >> (32 - d));
}
__device__ __forceinline__ void tf_round(uint32_t& x0, uint32_t& x1, int r) {
  x0 += x1; x1 = rotl32(x1, r); x1 ^= x0;
}
__device__ __forceinline__ void tf2x32(uint32_t k0, uint32_t k1,
                                       uint32_t& x0, uint32_t& x1) {
  uint32_t k2 = k0 ^ k1 ^ 0x1BD11BDAu;
  x0 += k0; x1 += k1;
  tf_round(x0, x1, 13); tf_round(x0, x1, 15); tf_round(x0, x1, 26); tf_round(x0, x1, 6);
  x0 += k1; x1 += k2 + 1u;
  tf_round(x0, x1, 17); tf_round(x0, x1, 29); tf_round(x0, x1, 16); tf_round(x0, x1, 24);
  x0 += k2; x1 += k0 + 2u;
  tf_round(x0, x1, 13); tf_round(x0, x1, 15); tf_round(x0, x1, 26); tf_round(x0, x1, 6);
  x0 += k0; x1 += k1 + 3u;
  tf_round(x0, x1, 17); tf_round(x0, x1, 29); tf_round(x0, x1, 16); tf_round(x0, x1, 24);
  x0 += k1; x1 += k2 + 4u;
  tf_round(x0, x1, 13); tf_round(x0, x1, 15); tf_round(x0, x1, 26); tf_round(x0, x1, 6);
  x0 += k2; x1 += k0 + 5u;
}

// ---------------- top-4 maintenance (lax.top_k tie rules) ----------------

__device__ __forceinline__ bool tk_better(float v, int i, float v2, int i2) {
  return (v > v2) || (v == v2 && i < i2);  // value desc, index asc
}
__device__ __forceinline__ void tk_insert(float (&tv)[K], int (&ti)[K],
                                          float v, int i) {
  if (!tk_better(v, i, tv[K - 1], ti[K - 1])) return;
  tv[K - 1] = v; ti[K - 1] = i;
#pragma unroll
  for (int q = K - 1; q > 0; --q) {
    if (tk_better(tv[q], ti[q], tv[q - 1], ti[q - 1])) {
      float fv = tv[q]; tv[q] = tv[q - 1]; tv[q - 1] = fv;
      int fi = ti[q]; ti[q] = ti[q - 1]; ti[q - 1] = fi;
    }
  }
}

// ---------------- kernel 0: RNG keys + zeroing ----------------

__global__ __launch_bounds__(THREADS) void k_init(uint32_t* __restrict__ keys2,
                                                  int* __restrict__ npos,
                                                  int* __restrict__ counts) {
  __shared__ uint32_t outw[16];
  const int t = threadIdx.x;
  if (t < 8) {  // jax.random.split(key(42), 8): threefry((0,42), pairs (i, i+8))
    uint32_t x0 = (uint32_t)t, x1 = (uint32_t)t + 8u;
    tf2x32(0u, 42u, x0, x1);
    outw[t] = x0; outw[t + 8] = x1;
  }
  __syncthreads();
  if (t < 16) keys2[t] = outw[t];   // keys2[b*2+j] == concat(out0,out1)[2b+j]
  if (t < B) npos[t] = 0;
  for (int i = t; i < B * G; i += THREADS) counts[i] = 0;
}

// ---------------- kernel 1: per-(b,g) top-4 IoU anchors + hq ----------------

__global__ __launch_bounds__(THREADS) void k_topk(
    const float* __restrict__ anchors, const float* __restrict__ gt_boxes,
    float* __restrict__ topv, int* __restrict__ topi, float* __restrict__ hq) {
  const int b = blockIdx.x >> 6;
  const int g = blockIdx.x & (G - 1);
  const int t = threadIdx.x;
  const float4 gtb =
      to_xyxy(reinterpret_cast<const float4*>(gt_boxes)[b * G + g]);
  const float4* anc = reinterpret_cast<const float4*>(anchors) + (size_t)b * A;

  float tv[K] = {-1e38f, -1e38f, -1e38f, -1e38f};
  int ti[K] = {0x7FFFFFFF, 0x7FFFFFFF, 0x7FFFFFFF, 0x7FFFFFFF};
  for (int a = t; a < A; a += THREADS) {
    __builtin_prefetch(anc + a + 4 * THREADS, 0, 1);  // global_prefetch_b8
    float v = iou_f(gtb, to_xyxy(anc[a]));
    tk_insert(tv, ti, v, a);
  }
  __shared__ float sv[THREADS][K];
  __shared__ int si[THREADS][K];
#pragma unroll
  for (int k = 0; k < K; ++k) { sv[t][k] = tv[k]; si[t][k] = ti[k]; }
  __syncthreads();
  for (int s = THREADS / 2; s > 0; s >>= 1) {
    if (t < s) {
#pragma unroll
      for (int k = 0; k < K; ++k) tk_insert(tv, ti, sv[t + s][k], si[t + s][k]);
#pragma unroll
      for (int k = 0; k < K; ++k) { sv[t][k] = tv[k]; si[t][k] = ti[k]; }
    }
    __syncthreads();
  }
  if (t == 0) {
#pragma unroll
    for (int k = 0; k < K; ++k) {
      topv[(b * G + g) * K + k] = tv[k];
      topi[(b * G + g) * K + k] = ti[k];
    }
    hq[b * G + g] = tv[0];
  }
}

// ---------------- kernel 2: per-anchor positivity + RNG bits ----------------

__global__ __launch_bounds__(THREADS) void k_assign(
    const float* __restrict__ anchors, const float* __restrict__ gt_boxes,
    const float* __restrict__ hq, const uint32_t* __restrict__ keys2,
    uint32_t* __restrict__ rq, int* __restrict__ npos) {
  __shared__ float4 sGT[G];   // 1 KB
  __shared__ float sHQ[G];    // 256 B
  const int b = blockIdx.y;
  const int t = threadIdx.x;

  // CDNA5 async copy: GT boxes (64x16B) + hq (16x16B) -> LDS
  uint32_t ldsGT = (uint32_t)(uintptr_t)(&sGT[0]);
  uint32_t ldsHQ = (uint32_t)(uintptr_t)(&sHQ[0]);
  uint64_t gsrc = (uint64_t)(uintptr_t)(gt_boxes + (size_t)b * G * 4);
  uint64_t hsrc = (uint64_t)(uintptr_t)(hq + (size_t)b * G);
  if (t < G) {
    asm volatile("global_load_async_to_lds_b128 %0, %1, off"
                 :: "v"(ldsGT + (uint32_t)t * 16u),
                    "v"(gsrc + (uint64_t)t * 16u)
                 : "memory");
  } else if (t < G + 16) {
    uint32_t u = (uint32_t)(t - G);
    asm volatile("global_load_async_to_lds_b128 %0, %1, off"
                 :: "v"(ldsHQ + u * 16u), "v"(hsrc + (uint64_t)u * 16u)
                 : "memory");
  }
  asm volatile("s_wait_asynccnt 0" ::: "memory");
  __syncthreads();
  if (t < G) sGT[t] = to_xyxy(sGT[t]);
  __syncthreads();

  const int a = blockIdx.x * THREADS + t;
  if (a >= A) return;
  float4 an =
      to_xyxy(reinterpret_cast<const float4*>(anchors)[(size_t)b * A + a]);
  float best = -1e38f;
  bool pos = false;
  for (int g = 0; g < G; ++g) {
    float v = iou_f(sGT[g], an);
    if (v > best) best = v;
    if (v == sHQ[g]) pos = true;  // low-quality match rule (exact equality)
  }
  if (best >= T_HIGH) pos = true;

  // r = uniform(key_b, (A,)): pair (a', a'+A/2), a' = a mod A/2
  uint32_t k0 = keys2[b * 2 + 0], k1 = keys2[b * 2 + 1];
  const uint32_t half = (uint32_t)(A / 2);
  uint32_t ap = (uint32_t)a;
  bool lohalf = ap < half;
  if (!lohalf) ap -= half;
  uint32_t x0 = ap, x1 = ap + half;
  tf2x32(k0, k1, x0, x1);
  uint32_t bits = lohalf ? x0 : x1;
  uint32_t q = bits >> 9;  // 23-bit mantissa; ordering/ties identical to r
  rq[(size_t)b * A + a] = pos ? q : 0xFFFFFFFFu;
  if (pos) atomicAdd(&npos[b], 1);
}

// ---------------- kernel 3: 128th-smallest (r, idx) key per image ----------------

__global__ __launch_bounds__(THREADS) void k_cutoff(
    const uint32_t* __restrict__ rq, const int* __restrict__ npos,
    unsigned long long* __restrict__ cutoff) {
  const int b = blockIdx.x;
  const int t = threadIdx.x;
  __shared__ int red[THREADS];
  const int np = npos[b];
  if (np <= MAX_POS) {  // all positives selected
    if (t == 0) cutoff[b] = 0xFFFFFFFE00000000ULL;  // > any positive, < sentinel
    return;
  }
  const uint32_t* r = rq + (size_t)b * A;
  unsigned long long lo = 0, hi = 0x0080000000000000ULL;  // > max positive key
  while (lo < hi) {
    unsigned long long mid = lo + ((hi - lo) >> 1);
    int c = 0;
    for (int a = t; a < A; a += THREADS) {
      unsigned long long key =
          ((unsigned long long)r[a] << 32) | (uint32_t)a;
      c += (key <= mid) ? 1 : 0;
    }
    red[t] = c;
    __syncthreads();
    for (int s = THREADS / 2; s > 0; s >>= 1) {
      if (t < s) red[t] += red[t + s];
      __syncthreads();
    }
    int total = red[0];
    __syncthreads();
    if (total >= MAX_POS) hi = mid; else lo = mid + 1;
  }
  if (t == 0) cutoff[b] = lo;  // exactly 128 unique keys <= lo
}

// ---------------- kernel 4: per-GT counts of selected positives ----------------

__global__ __launch_bounds__(THREADS) void k_counts(
    const float* __restrict__ anchors, const float* __restrict__ gt_boxes,
    const uint32_t* __restrict__ rq, const unsigned long long* __restrict__ cutoff,
    int* __restrict__ counts) {
  const int b = blockIdx.y;
  const int a = blockIdx.x * THREADS + threadIdx.x;
  if (a >= A) return;
  unsigned long long key =
      ((unsigned long long)rq[(size_t)b * A + a] << 32) | (uint32_t)a;
  if (key > cutoff[b]) return;  // <=128 anchors per image pass this
  float4 an =
      to_xyxy(reinterpret_cast<const float4*>(anchors)[(size_t)b * A + a]);
  const float4* gt = reinterpret_cast<const float4*>(gt_boxes) + (size_t)b * G;
  float best = -1e38f;
  int bi = 0;
  for (int g = 0; g < G; ++g) {
    float v = iou_f(to_xyxy(gt[g]), an);
    if (v > best) { best = v; bi = g; }  // jnp.argmax: first occurrence
  }
  atomicAdd(&counts[b * G + bi], 1);
}

// ---------------- kernel 5: emit (pr, gt_i, valid, scores) as floats ----------------

__global__ __launch_bounds__(THREADS) void k_emit(
    const float* __restrict__ topv, const int* __restrict__ topi,
    const int* __restrict__ counts, float* __restrict__ out) {
  const int idx = blockIdx.x * THREADS + threadIdx.x;
  if (idx >= B * G * K) return;
  const int k = idx & (K - 1);
  const int g = (idx / K) & (G - 1);
  const int b = idx / (G * K);
  const int c = counts[b * G + g];
  const int m = c < K ? c : K;
  const bool val = k < m;
  const int GK = G * K;
  const int o = b * GK + g * K + k;
  out[0 * B * GK + o] = val ? (float)topi[(b * G + g) * K + k] : -1.0f;
  out[1 * B * GK + o] = val ? (float)g : -1.0f;
  out[2 * B * GK + o] = val ? 1.0f : 0.0f;
  out[3 * B * GK + o] = val ? topv[(b * G + g) * K + k] : 0.0f;
}

// ---------------- launcher ----------------

extern "C" void kernel_launch(void* const* d_in, const int* in_sizes, int n_in,
                              void* d_out, int out_size, void* d_ws,
                              size_t ws_size, hipStream_t stream) {
  (void)in_sizes; (void)n_in; (void)out_size; (void)ws_size;
  const float* anchors = (const float*)d_in[0];    // [B, A, 4] cxcywh
  const float* gt_boxes = (const float*)d_in[1];   // [B, G, 4] cxcywh
  float* out = (float*)d_out;                      // 4 * [B, G*K] floats

  // workspace layout (~6.42 MB total)
  char* ws = (char*)d_ws;
  uint32_t* keys2 = (uint32_t*)(ws + 0);                    // [B][2]
  int* npos = (int*)(ws + 64);                              // [B]
  unsigned long long* cutoff = (unsigned long long*)(ws + 96);  // [B]
  int* counts = (int*)(ws + 160);                           // [B][G]
  float* topv = (float*)(ws + 2208);                        // [B][G][K]
  int* topi = (int*)(ws + 10400);                           // [B][G][K]
  float* hq = (float*)(ws + 18592);                         // [B][G]
  uint32_t* rq = (uint32_t*)(ws + 20736);                   // [B][A]

  k_init<<<1, THREADS, 0, stream>>>(keys2, npos, counts);
  k_topk<<<B * G, THREADS, 0, stream>>>(anchors, gt_boxes, topv, topi, hq);
  dim3 gridA(ACH, B);
  k_assign<<<gridA, THREADS, 0, stream>>>(anchors, gt_boxes, hq, keys2, rq, npos);
  k_cutoff<<<B, THREADS, 0, stream>>>(rq, npos, cutoff);
  k_counts<<<gridA, THREADS, 0, stream>>>(anchors, gt_boxes, rq, cutoff, counts);
  k_emit<<<(B * G * K + THREADS - 1) / THREADS, THREADS, 0, stream>>>(
      topv, topi, counts, out);
}